// SurfaceGraphCommunication_71485435675228
// MI455X (gfx1250) — compile-verified
//
#include <hip/hip_runtime.h>
#include <math.h>

// ---------------------------------------------------------------------------
// Problem constants (match reference)
// ---------------------------------------------------------------------------
#define N_S 30000
#define N_G 3000
#define DIM 128
#define TOPK 16
#define INV_2SIG2 (1.0f / 12.5f)   // 1/(2*2.5^2)

typedef __attribute__((ext_vector_type(2))) float v2f;
typedef __attribute__((ext_vector_type(4))) float v4f;
typedef __attribute__((ext_vector_type(8))) float v8f;

// CDNA5 async global->LDS copy path (guarded; falls back to vector loads)
#ifndef HAVE_GFX1250_ASYNC
#if defined(__has_builtin)
#if __has_builtin(__builtin_amdgcn_global_load_async_to_lds_b128) && \
    __has_builtin(__builtin_amdgcn_s_wait_asynccnt)
#define HAVE_GFX1250_ASYNC 1
#endif
#endif
#endif
#ifndef HAVE_GFX1250_ASYNC
#define HAVE_GFX1250_ASYNC 0
#endif

#if HAVE_GFX1250_ASYNC
// From the compiler diagnostic: param 1 is
//   '__attribute__((__vector_size__(4 * sizeof(int)))) int __device__ *'
// i.e. int4 in addrspace(1); LDS side is the matching int4 in addrspace(3).
typedef __attribute__((__vector_size__(4 * sizeof(int)))) int i4vec;
typedef __attribute__((address_space(1))) i4vec* gv4i_p;
typedef __attribute__((address_space(3))) i4vec* lv4i_p;
#endif

// ---------------------------------------------------------------------------
// Small helper kernels
// ---------------------------------------------------------------------------
__global__ void row_norm3_kernel(const float* __restrict__ pos, float* __restrict__ out, int n) {
    int i = blockIdx.x * blockDim.x + threadIdx.x;
    if (i < n) {
        float x = pos[3 * i + 0], y = pos[3 * i + 1], z = pos[3 * i + 2];
        out[i] = x * x + y * y + z * z;
    }
}

__global__ void zero_f32_kernel(float* __restrict__ p, int n) {
    int i = blockIdx.x * blockDim.x + threadIdx.x;
    if (i < n) p[i] = 0.0f;
}

// ---------------------------------------------------------------------------
// kNN top-16: one 256-thread workgroup per graph node.
// Per-thread branchless sorted top-16 in registers, then 16 rounds of LDS
// argmin reduction (ascending d2, ties -> lower index, matching lax.top_k).
// ---------------------------------------------------------------------------
__global__ __launch_bounds__(256) void knn_topk_kernel(
    const float* __restrict__ pos_s, const float* __restrict__ pos_g,
    const float* __restrict__ ssn, const float* __restrict__ sgn,
    int* __restrict__ idx_out, float* __restrict__ w_out) {

    __shared__ float cd[256 * TOPK];
    __shared__ int   ci[256 * TOPK];
    __shared__ float rd[256];
    __shared__ int   ri[256];
    __shared__ int   rs[256];

    const int g = blockIdx.x;
    const int t = threadIdx.x;

    const float gx = pos_g[3 * g + 0];
    const float gy = pos_g[3 * g + 1];
    const float gz = pos_g[3 * g + 2];
    const float gn = sgn[g];

    float bd[TOPK];
    int   bi[TOPK];
#pragma unroll
    for (int j = 0; j < TOPK; ++j) { bd[j] = 3.4e38f; bi[j] = 0x7fffffff; }

    for (int i = t; i < N_S; i += 256) {
        float sx = pos_s[3 * i + 0], sy = pos_s[3 * i + 1], sz = pos_s[3 * i + 2];
        float d = gn + ssn[i] - 2.0f * (gx * sx + gy * sy + gz * sz);
        if (d < bd[TOPK - 1]) {
#pragma unroll
            for (int j = TOPK - 1; j >= 1; --j) {
                if (d < bd[j]) {
                    bool shift = d < bd[j - 1];
                    bd[j] = shift ? bd[j - 1] : d;
                    bi[j] = shift ? bi[j - 1] : i;
                }
            }
            if (d < bd[0]) { bd[0] = d; bi[0] = i; }
        }
    }

#pragma unroll
    for (int j = 0; j < TOPK; ++j) { cd[t * TOPK + j] = bd[j]; ci[t * TOPK + j] = bi[j]; }
    __syncthreads();

    for (int r = 0; r < TOPK; ++r) {
        float md = 3.4e38f; int mi = 0x7fffffff; int ms = 0;
#pragma unroll
        for (int j = 0; j < TOPK; ++j) {
            float dv = cd[t * TOPK + j]; int iv = ci[t * TOPK + j];
            if (dv < md || (dv == md && iv < mi)) { md = dv; mi = iv; ms = t * TOPK + j; }
        }
        rd[t] = md; ri[t] = mi; rs[t] = ms;
        __syncthreads();
        for (int off = 128; off > 0; off >>= 1) {
            if (t < off) {
                float dv = rd[t + off]; int iv = ri[t + off];
                if (dv < rd[t] || (dv == rd[t] && iv < ri[t])) {
                    rd[t] = dv; ri[t] = iv; rs[t] = rs[t + off];
                }
            }
            __syncthreads();
        }
        if (t == 0) {
            idx_out[g * TOPK + r] = ri[0];
            w_out[g * TOPK + r]   = expf(-rd[0] * INV_2SIG2);
            cd[rs[0]] = 3.4e38f;
            ci[rs[0]] = 0x7fffffff;
        }
        __syncthreads();
    }
}

// ---------------------------------------------------------------------------
// Dense fp32 GEMM via V_WMMA_F32_16X16X4_F32 with LDS-staged B panel.
//   C[M x 128] = op(A)[M x K] * B[K x 128], row-major, K in {128, 256}.
//   TWO_A: A = concat([A0 | A1]) along K, each half row stride 128.
//   RELU : max(0, .) before store.
// Block = 256 threads (8 wave32), computes a 128x128 C block:
//   wave w owns M-tile (blockIdx*128 + w*16 .. +15) across all 8 N-tiles.
// B is staged in 128-row chunks (64KB LDS) via CDNA5 async-copy when
// available (global_load_async_to_lds_b128 + s_wait_asynccnt).
// Fragment layouts per CDNA5 ISA fp32 tables:
//   A 16x4: lane<16 -> row=lane, k={ka,ka+1}; lane>=16 -> k={ka+2,ka+3}
//   C/D 16x16: VGPR r -> row r + 8*(lane>=16), col = lane&15
// ---------------------------------------------------------------------------
template <int TWO_A, int RELU, int K>
__global__ __launch_bounds__(256) void gemm_wmma_kernel(
    const float* __restrict__ A0, const float* __restrict__ A1,
    const float* __restrict__ B, float* __restrict__ C, int M) {

    __shared__ float bsh[128 * 128];          // one 128-row chunk of B (64KB)

    const int tid  = threadIdx.x;
    const int lane = tid & 31;
    const int wv   = tid >> 5;                // 0..7 -> M-tile within block
    const int half = lane >> 4;
    const int r16  = lane & 15;

    const int m  = blockIdx.x * 128 + wv * 16 + r16;
    const int mc = (m < M) ? m : (M - 1);     // clamp: junk rows never stored

    const v8f zero = {};
    v8f acc[8];
#pragma unroll
    for (int nt = 0; nt < 8; ++nt) acc[nt] = zero;

#pragma unroll
    for (int kc = 0; kc < K; kc += 128) {
        __syncthreads();                      // protect LDS reuse across chunks
        // ---- stage B[kc..kc+127][0..127] into LDS ----
        const float* Bsrc = B + (long)kc * 128;
#if HAVE_GFX1250_ASYNC
        for (int i = tid * 4; i < 128 * 128; i += 256 * 4) {
            __builtin_amdgcn_global_load_async_to_lds_b128(
                (gv4i_p)(Bsrc + i), (lv4i_p)(bsh + i), 0, 0);
        }
        __builtin_amdgcn_s_wait_asynccnt(0);
#else
        for (int i = tid * 4; i < 128 * 128; i += 256 * 4) {
            *(v4f*)(bsh + i) = *(const v4f*)(Bsrc + i);
        }
#endif
        __syncthreads();

        // ---- A row pointer for this chunk ----
        const float* Arow;
        if (TWO_A) Arow = (kc == 0 ? A0 : A1) + (long)mc * 128;
        else       Arow = A0 + (long)mc * K + kc;

#pragma unroll 4
        for (int kk = 0; kk < 128; kk += 4) {
            const int ka = kk + half * 2;     // ka, ka+1 stay within this 4-group
            const v2f a = *(const v2f*)(Arow + ka);
#pragma unroll
            for (int nt = 0; nt < 8; ++nt) {
                v2f b;
                b[0] = bsh[ka * 128 + nt * 16 + r16];
                b[1] = bsh[(ka + 1) * 128 + nt * 16 + r16];
                acc[nt] = __builtin_amdgcn_wmma_f32_16x16x4_f32(
                    false, a, false, b, (short)0, acc[nt], false, false);
            }
        }
    }

    // ---- store 16x128 strip ----
#pragma unroll
    for (int nt = 0; nt < 8; ++nt) {
#pragma unroll
        for (int r = 0; r < 8; ++r) {
            const int mr = blockIdx.x * 128 + wv * 16 + r + half * 8;
            if (mr < M) {
                float v = acc[nt][r];
                if (RELU) v = fmaxf(v, 0.0f);
                C[(long)mr * 128 + nt * 16 + r16] = v;
            }
        }
    }
}

// ---------------------------------------------------------------------------
// Edge kernels: graph->surface scatter-add, surface->graph gather-sum.
// ---------------------------------------------------------------------------
__global__ __launch_bounds__(128) void scatter_gs_kernel(
    const float* __restrict__ hgW, const int* __restrict__ idx,
    const float* __restrict__ w, float* __restrict__ xs_out) {
    const int g = blockIdx.x, t = threadIdx.x;
    const float v = hgW[g * DIM + t];
#pragma unroll
    for (int k = 0; k < TOPK; ++k) {
        const int   row = idx[g * TOPK + k];
        const float wk  = w[g * TOPK + k];
        atomicAdd(&xs_out[(long)row * DIM + t], v * wk);
    }
}

__global__ __launch_bounds__(128) void gather_sg_kernel(
    const float* __restrict__ hsW, const int* __restrict__ idx,
    const float* __restrict__ w, float* __restrict__ xg_out) {
    const int g = blockIdx.x, t = threadIdx.x;
    float acc = 0.0f;
#pragma unroll
    for (int k = 0; k < TOPK; ++k) {
        const int   row = idx[g * TOPK + k];
        const float wk  = w[g * TOPK + k];
        acc += hsW[(long)row * DIM + t] * wk;
    }
    xg_out[g * DIM + t] = acc;
}

// ---------------------------------------------------------------------------
// Host-side launcher
// ---------------------------------------------------------------------------
extern "C" void kernel_launch(void* const* d_in, const int* in_sizes, int n_in,
                              void* d_out, int out_size, void* d_ws, size_t ws_size,
                              hipStream_t stream) {
    (void)in_sizes; (void)n_in; (void)out_size; (void)ws_size;

    const float* xs       = (const float*)d_in[0];
    const float* xg       = (const float*)d_in[1];
    const float* pos_s    = (const float*)d_in[2];
    const float* pos_g    = (const float*)d_in[3];
    const float* W_s_pre  = (const float*)d_in[4];
    const float* W_g_pre  = (const float*)d_in[5];
    const float* W_gs     = (const float*)d_in[6];
    const float* W_sg     = (const float*)d_in[7];
    const float* W_s_post = (const float*)d_in[8];
    const float* W_g_post = (const float*)d_in[9];
    // d_in[10] = k (always 16)

    float* out_s = (float*)d_out;                 // xs_new (30000 x 128)
    float* out_g = out_s + (long)N_S * DIM;       // xg_new (3000 x 128)

    // workspace layout
    float* ws    = (float*)d_ws;
    float* hs    = ws;                            // N_S * D
    float* hsW   = hs    + (long)N_S * DIM;       // N_S * D
    float* xs_o  = hsW   + (long)N_S * DIM;       // N_S * D
    float* hg    = xs_o  + (long)N_S * DIM;       // N_G * D
    float* hgW   = hg    + (long)N_G * DIM;       // N_G * D
    float* xg_o  = hgW   + (long)N_G * DIM;       // N_G * D
    float* ssn   = xg_o  + (long)N_G * DIM;       // N_S
    float* sgn   = ssn   + N_S;                   // N_G
    int*   idx   = (int*)(sgn + N_G);             // N_G * 16
    float* wgt   = (float*)(idx + N_G * TOPK);    // N_G * 16

    const int gbS = (N_S + 127) / 128;            // GEMM blocks (128 rows each)
    const int gbG = (N_G + 127) / 128;

    // 1) position norms + kNN/top-16 with RBF weights
    row_norm3_kernel<<<(N_S + 255) / 256, 256, 0, stream>>>(pos_s, ssn, N_S);
    row_norm3_kernel<<<(N_G + 255) / 256, 256, 0, stream>>>(pos_g, sgn, N_G);
    knn_topk_kernel<<<N_G, 256, 0, stream>>>(pos_s, pos_g, ssn, sgn, idx, wgt);

    // 2) pre blocks (fp32 WMMA GEMMs)
    gemm_wmma_kernel<0, 0, 128><<<gbS, 256, 0, stream>>>(xs, nullptr, W_s_pre, hs, N_S);
    gemm_wmma_kernel<0, 0, 128><<<gbG, 256, 0, stream>>>(xg, nullptr, W_g_pre, hg, N_G);

    // 3) message transforms
    gemm_wmma_kernel<0, 0, 128><<<gbS, 256, 0, stream>>>(hs, nullptr, W_sg, hsW, N_S);
    gemm_wmma_kernel<0, 0, 128><<<gbG, 256, 0, stream>>>(hg, nullptr, W_gs, hgW, N_G);

    // 4) sparse exchange: graph->surface scatter-add, surface->graph gather-sum
    zero_f32_kernel<<<((long)N_S * DIM + 255) / 256, 256, 0, stream>>>(xs_o, N_S * DIM);
    scatter_gs_kernel<<<N_G, DIM, 0, stream>>>(hgW, idx, wgt, xs_o);
    gather_sg_kernel<<<N_G, DIM, 0, stream>>>(hsW, idx, wgt, xg_o);

    // 5) post blocks: relu(concat(pre, msg) @ W_post), fused concat + ReLU
    gemm_wmma_kernel<1, 1, 256><<<gbS, 256, 0, stream>>>(hs, xs_o, W_s_post, out_s, N_S);
    gemm_wmma_kernel<1, 1, 256><<<gbG, 256, 0, stream>>>(hg, xg_o, W_g_post, out_g, N_G);
}